// RtrlCell_26362509263066
// MI455X (gfx1250) — compile-verified
//
#include <hip/hip_runtime.h>
#include <math.h>

typedef __attribute__((ext_vector_type(2))) float v2f;
typedef __attribute__((ext_vector_type(8))) float v8f;
typedef __attribute__((ext_vector_type(4))) int   v4i;

#define HH   128
#define TC   16          // columns per WMMA tile
#define NT   64          // tiles per block -> 1024-column slab
#define SLAB (NT * TC)

#define AS1 __attribute__((address_space(1)))
#define AS3 __attribute__((address_space(3)))

// async copy of one 16B chunk: global -> LDS, no VGPR round trip (ASYNCcnt)
__device__ __forceinline__ void async_cp16(const float* g, float* l) {
    __builtin_amdgcn_global_load_async_to_lds_b128(
        (AS1 v4i*)g, (AS3 v4i*)l, 0, 0);
}

// wave waits until at most N async-LDS ops outstanding
#define WAIT_ASYNC(n) asm volatile("s_wait_asynccnt " #n ::: "memory")

// new_S[b,k,c] = sum_n (R[n,k] * (1 - ph[b,n]^2)) * S[b,n,c]  (+ diag term at j==k)
// One launch covers S_R (z=0) and S_W (z=1).
__global__ __launch_bounds__(256) void rtrl_gemm_kernel(
    const float* __restrict__ R,
    const float* __restrict__ prev_h,
    const float* __restrict__ x,
    const float* __restrict__ S_R,
    const float* __restrict__ S_W,
    float* __restrict__ out_SR,
    float* __restrict__ out_SW)
{
    const int b   = blockIdx.y;
    const int sel = blockIdx.z;
    const int c0  = blockIdx.x * SLAB;

    const size_t plane = (size_t)HH * HH * HH;          // 2,097,152 floats per batch
    const float* S  = (sel ? S_W : S_R) + (size_t)b * plane;
    float*       O  = (sel ? out_SW : out_SR) + (size_t)b * plane;
    const float* dv = (sel ? x : prev_h) + b * HH;      // diagonal add vector (over i)

    // Plain row-major tile: Bt[buf][n][col]; async b128 writes one 16B row chunk.
    // Fragment reads (n0,m),(n0+1,m): lanes0-15 banks m/16+m, lanes16-31 banks 32+m/48+m.
    __shared__ float Bt[3][HH][TC];
    __shared__ float vdiag[HH];

    const int tid  = threadIdx.x;
    const int wave = tid >> 5;
    const int lane = tid & 31;
    const int half = lane >> 4;      // K-group: lanes 0-15 -> K=0,1 ; 16-31 -> K=2,3
    const int m    = lane & 15;
    const int krow = wave * 16 + m;  // output row (M index) this lane feeds into A

    const int quad = tid & 3;        // which 16B chunk of the 64B tile row
    const int r0   = tid >> 2;       // this thread copies rows r0 and r0+64

    // ---- prologue: async-stage tiles 0 and 1 (2 instrs per wave per tile)
    {
        const float* g = S + (size_t)r0 * (HH * HH) + c0 + quad * 4;
        async_cp16(g,                          &Bt[0][r0][quad * 4]);
        async_cp16(g + (size_t)64 * (HH * HH), &Bt[0][r0 + 64][quad * 4]);
        async_cp16(g + TC,                          &Bt[1][r0][quad * 4]);
        async_cp16(g + TC + (size_t)64 * (HH * HH), &Bt[1][r0 + 64][quad * 4]);
    }

    // ---- Build A fragments in registers: a[s] holds A[krow][4s+2*half .. +1]
    const float* phb = prev_h + b * HH;
    v2f a[32];
#pragma unroll
    for (int s = 0; s < 32; ++s) {
        const int n0 = 4 * s + 2 * half;
        const float p0 = phb[n0];
        const float p1 = phb[n0 + 1];
        a[s].x = R[n0 * HH + krow] * (1.0f - p0 * p0);
        a[s].y = R[(n0 + 1) * HH + krow] * (1.0f - p1 * p1);
    }

    if (tid < HH) vdiag[tid] = dv[tid];

    for (int t = 0; t < NT; ++t) {
        // retire tile t's copies (2 per wave still in flight for t+1, if any)
        if (t + 1 < NT) { WAIT_ASYNC(2); } else { WAIT_ASYNC(0); }
        __syncthreads();   // publish tile t to all waves; all waves done reading buf (t+2)%3

        // kick off tile t+2 into the buffer consumed at iteration t-1
        if (t + 2 < NT) {
            const int   cn = c0 + (t + 2) * TC;
            const int   nb = (t + 2) % 3;
            const float* g = S + (size_t)r0 * (HH * HH) + cn + quad * 4;
            async_cp16(g,                          &Bt[nb][r0][quad * 4]);
            async_cp16(g + (size_t)64 * (HH * HH), &Bt[nb][r0 + 64][quad * 4]);
        }

        // ---- 32 V_WMMA_F32_16X16X4_F32, two interleaved accumulator chains
        const float* tb = &Bt[t % 3][0][0];
        v8f acc0 = {};
        v8f acc1 = {};
#pragma unroll
        for (int s = 0; s < 32; s += 2) {
            const int n0 = 4 * s + 2 * half;
            v2f bf0, bf1;
            bf0.x = tb[n0 * TC + m];
            bf0.y = tb[(n0 + 1) * TC + m];
            bf1.x = tb[(n0 + 4) * TC + m];
            bf1.y = tb[(n0 + 5) * TC + m];
            acc0 = __builtin_amdgcn_wmma_f32_16x16x4_f32(
                false, a[s],     false, bf0, (short)0, acc0, false, false);
            acc1 = __builtin_amdgcn_wmma_f32_16x16x4_f32(
                false, a[s + 1], false, bf1, (short)0, acc1, false, false);
        }

        // ---- store D; VGPR v: lanes0-15 -> M=v, lanes16-31 -> M=v+8, N=m
        const int   c    = c0 + t * TC + m;
        const int   j    = c & (HH - 1);
        const float dval = vdiag[c >> 7];
#pragma unroll
        for (int v = 0; v < 8; ++v) {
            const int kk  = wave * 16 + v + 8 * half;
            float     val = acc0[v] + acc1[v];
            if (j == kk) val += dval;                         // fused diagonal term
            __builtin_nontemporal_store(val, &O[(size_t)kk * (HH * HH) + c]);
        }
        // no trailing barrier: next iteration's wait+barrier licenses buffer reuse
    }
}

// curr_h[b,j] = tanh(x@Wi + bi + prev_h@R)
__global__ void rtrl_h_kernel(const float* __restrict__ x,
                              const float* __restrict__ prev_h,
                              const float* __restrict__ Wi,
                              const float* __restrict__ bi,
                              const float* __restrict__ R,
                              float* __restrict__ curr_h)
{
    const int b = blockIdx.x;
    const int j = threadIdx.x;
    float s = bi[j];
    for (int i = 0; i < HH; ++i) s += x[b * HH + i] * Wi[i * HH + j];
    for (int n = 0; n < HH; ++n) s += prev_h[b * HH + n] * R[n * HH + j];
    curr_h[b * HH + j] = tanhf(s);
}

// new_S_B[b,k,j] = sum_n R[n,k]*(1-ph^2)*S_B[b,n,j] + (k==j)
__global__ void rtrl_sb_kernel(const float* __restrict__ prev_h,
                               const float* __restrict__ R,
                               const float* __restrict__ S_B,
                               float* __restrict__ out_SB)
{
    const int k = blockIdx.x;
    const int b = blockIdx.y;
    const int j = threadIdx.x;
    float s = (j == k) ? 1.0f : 0.0f;
    for (int n = 0; n < HH; ++n) {
        const float ph = prev_h[b * HH + n];
        s += R[n * HH + k] * (1.0f - ph * ph) * S_B[((size_t)b * HH + n) * HH + j];
    }
    out_SB[((size_t)b * HH + k) * HH + j] = s;
}

extern "C" void kernel_launch(void* const* d_in, const int* in_sizes, int n_in,
                              void* d_out, int out_size, void* d_ws, size_t ws_size,
                              hipStream_t stream) {
    const float* x      = (const float*)d_in[0];
    const float* prev_h = (const float*)d_in[1];
    const float* S_R    = (const float*)d_in[2];
    const float* S_W    = (const float*)d_in[3];
    const float* S_B    = (const float*)d_in[4];
    const float* Wi     = (const float*)d_in[5];
    const float* bi     = (const float*)d_in[6];
    const float* R      = (const float*)d_in[7];

    float* out    = (float*)d_out;
    float* curr_h = out;                                   // 16*128
    float* out_SR = curr_h + 16 * HH;                      // 16*128^3
    float* out_SW = out_SR + (size_t)16 * HH * HH * HH;
    float* out_SB = out_SW + (size_t)16 * HH * HH * HH;    // 16*128*128

    // big einsums: 16 slabs x 16 batches x {S_R, S_W}
    dim3 grid(HH * HH * HH / SLAB / HH, 16, 2);            // (16,16,2)
    rtrl_gemm_kernel<<<grid, 256, 0, stream>>>(R, prev_h, x, S_R, S_W, out_SR, out_SW);

    rtrl_h_kernel<<<16, HH, 0, stream>>>(x, prev_h, Wi, bi, R, curr_h);
    rtrl_sb_kernel<<<dim3(HH, 16), HH, 0, stream>>>(prev_h, R, S_B, out_SB);
}